// GNN_13606456394522
// MI455X (gfx1250) — compile-verified
//
#include <hip/hip_runtime.h>
#include <math.h>

typedef __attribute__((ext_vector_type(2))) float v2f;
typedef __attribute__((ext_vector_type(4))) float v4f;
typedef __attribute__((ext_vector_type(8))) float v8f;

#define N_NODES 50000
#define N_EDGES 1600000
#define XDIM    28
#define NTILES  (N_EDGES / 16)   // 100000, exact
#define WPB     8                // waves per block (wave32)
#define TPB     (WPB * 32)
#define FSTR    36               // LDS row stride (floats): conflict-free + 16B aligned
#define BN_EPS  1e-5f

// ---- order-preserving float<->uint encoding for atomic max ----
__device__ __forceinline__ unsigned encf(float x) {
    unsigned u = __float_as_uint(x);
    return (u & 0x80000000u) ? ~u : (u | 0x80000000u);
}
__device__ __forceinline__ float decf(unsigned u) {
    return (u & 0x80000000u) ? __uint_as_float(u & 0x7FFFFFFFu)
                             : __uint_as_float(~u);
}
// intra-wave LDS producer->consumer fence (no block barrier needed)
__device__ __forceinline__ void wave_lds_fence() {
    asm volatile("s_wait_dscnt 0" ::: "memory");
}

__device__ __forceinline__ float angle3(float ax, float ay, float az,
                                        float bx, float by, float bz) {
    float cx = ay * bz - az * by;
    float cy = az * bx - ax * bz;
    float cz = ax * by - ay * bx;
    float cn = sqrtf(cx * cx + cy * cy + cz * cz);
    float dt = ax * bx + ay * by + az * bz;
    return atan2f(cn, dt);
}

// One 16(M)x32(N) tile of  D = A(16x32) * W^T + bias  using v_wmma_f32_16x16x4_f32.
// Fragment layout (wave32): lane = hf*16+lm.
//   A step k8: v2f = { f[lm][4*k8 + 2*hf], f[lm][4*k8 + 2*hf + 1] }
//   B step k8: v2f = { W[n][4*k8 + 2*hf], W[n][4*k8 + 2*hf + 1] },  n = t*16+lm
//   C/D      : v8f, element r -> row (r + hf*8), channel n
__device__ __forceinline__ void mlp_layer(const float* ft, const float* Wl,
                                          const float* bias, int lm, int hf,
                                          v8f out[2]) {
    v2f a[8];
#pragma unroll
    for (int k8 = 0; k8 < 8; ++k8)
        a[k8] = *(const v2f*)(ft + lm * FSTR + k8 * 4 + hf * 2);
#pragma unroll
    for (int t = 0; t < 2; ++t) {
        const int n = t * 16 + lm;
        const float bb = bias[n];
        v8f c = {bb, bb, bb, bb, bb, bb, bb, bb};
#pragma unroll
        for (int k8 = 0; k8 < 8; ++k8) {
            v2f bv = *(const v2f*)(Wl + n * FSTR + k8 * 4 + hf * 2);
            c = __builtin_amdgcn_wmma_f32_16x16x4_f32(
                    false, a[k8], false, bv, (short)0, c, false, false);
        }
        out[t] = c;
    }
}

template <int PASS>
__global__ __launch_bounds__(TPB) void ppf_pass(
    const float* __restrict__ x, const float* __restrict__ pos,
    const float* __restrict__ nrm,
    const int* __restrict__ esrc, const int* __restrict__ edst,
    const float* __restrict__ W1, const float* __restrict__ b1,
    const float* __restrict__ g1, const float* __restrict__ be1,
    const float* __restrict__ W2, const float* __restrict__ b2,
    const float* __restrict__ g2, const float* __restrict__ be2,
    const float* __restrict__ W3, const float* __restrict__ b3,
    float* __restrict__ statOut,        // this pass's [sum32|sumsq32]
    const float* __restrict__ stat1,    // pass-1 sums (PASS>=2)
    const float* __restrict__ stat2,    // pass-2 sums (PASS==3)
    unsigned* __restrict__ acc)         // encoded segment-max (PASS==3)
{
    __shared__ float sW[3][32 * FSTR];
    __shared__ float sBias[3][32];
    __shared__ float tileBuf[WPB][16 * FSTR];
    __shared__ int   sdst[WPB][16];
    __shared__ float ssum[32], ssq[32];

    const int tid  = threadIdx.x;
    const int lane = tid & 31;
    const int wv   = tid >> 5;
    const int hf   = lane >> 4;
    const int lm   = lane & 15;

    // stage weights/biases in LDS (row stride FSTR)
    {
        const int nW = (PASS == 1) ? 1 : ((PASS == 2) ? 2 : 3);
        for (int w = 0; w < nW; ++w) {
            const float* Wp = (w == 0) ? W1 : ((w == 1) ? W2 : W3);
            for (int i = tid; i < 32 * 32; i += TPB)
                sW[w][(i >> 5) * FSTR + (i & 31)] = Wp[i];
        }
        if (tid < 32) {
            sBias[0][tid] = b1[tid];
            if (PASS >= 2) sBias[1][tid] = b2[tid];
            if (PASS == 3) sBias[2][tid] = b3[tid];
            if (PASS <= 2) { ssum[tid] = 0.f; ssq[tid] = 0.f; }
        }
    }
    __syncthreads();

    // per-lane BN affine params derived from global sums (biased variance)
    float sc1[2], sh1[2], sc2[2], sh2[2];
    if (PASS >= 2) {
#pragma unroll
        for (int t = 0; t < 2; ++t) {
            int c = t * 16 + lm;
            float m = stat1[c] * (1.f / N_EDGES);
            float v = stat1[32 + c] * (1.f / N_EDGES) - m * m;
            float s = g1[c] * rsqrtf(v + BN_EPS);
            sc1[t] = s; sh1[t] = be1[c] - m * s;
        }
    }
    if (PASS == 3) {
#pragma unroll
        for (int t = 0; t < 2; ++t) {
            int c = t * 16 + lm;
            float m = stat2[c] * (1.f / N_EDGES);
            float v = stat2[32 + c] * (1.f / N_EDGES) - m * m;
            float s = g2[c] * rsqrtf(v + BN_EPS);
            sc2[t] = s; sh2[t] = be2[c] - m * s;
        }
    }

    float accS[2] = {0.f, 0.f}, accQ[2] = {0.f, 0.f};
    float* ft = tileBuf[wv];

    for (int tIdx = blockIdx.x * WPB + wv; tIdx < NTILES; tIdx += gridDim.x * WPB) {
        // ---- build 16x32 feature tile: [ x[src] (28) | ppf (4) ] ----
        const int e    = tIdx * 16 + lm;
        const int sIdx = esrc[e];
        const int dIdx = edst[e];
        const float* xr = x + sIdx * XDIM;     // rows are 16B aligned (28*4=112)
        if (hf == 0) {
#pragma unroll
            for (int c4 = 0; c4 < 4; ++c4)
                *(v4f*)(ft + lm * FSTR + c4 * 4) = *(const v4f*)(xr + c4 * 4);
            sdst[wv][lm] = dIdx;
        } else {
#pragma unroll
            for (int c4 = 0; c4 < 3; ++c4)
                *(v4f*)(ft + lm * FSTR + 16 + c4 * 4) = *(const v4f*)(xr + 16 + c4 * 4);
            float px = pos[3 * sIdx + 0] - pos[3 * dIdx + 0];
            float py = pos[3 * sIdx + 1] - pos[3 * dIdx + 1];
            float pz = pos[3 * sIdx + 2] - pos[3 * dIdx + 2];
            float nix = nrm[3 * dIdx + 0], niy = nrm[3 * dIdx + 1], niz = nrm[3 * dIdx + 2];
            float njx = nrm[3 * sIdx + 0], njy = nrm[3 * sIdx + 1], njz = nrm[3 * sIdx + 2];
            ft[lm * FSTR + 28] = sqrtf(px * px + py * py + pz * pz);
            ft[lm * FSTR + 29] = angle3(nix, niy, niz, px, py, pz);
            ft[lm * FSTR + 30] = angle3(njx, njy, njz, px, py, pz);
            ft[lm * FSTR + 31] = angle3(nix, niy, niz, njx, njy, njz);
        }
        wave_lds_fence();

        // ---- layer 1 ----
        v8f h[2];
        mlp_layer(ft, sW[0], sBias[0], lm, hf, h);
        if (PASS == 1) {
#pragma unroll
            for (int t = 0; t < 2; ++t) {
                float ls = 0.f, lq = 0.f;
#pragma unroll
                for (int r = 0; r < 8; ++r) { float v = h[t][r]; ls += v; lq += v * v; }
                accS[t] += ls; accQ[t] += lq;
            }
            continue;
        }
        // BN1 + ReLU, write r1 back into the tile (A-frags already in regs)
#pragma unroll
        for (int t = 0; t < 2; ++t) {
            int n = t * 16 + lm;
#pragma unroll
            for (int r = 0; r < 8; ++r)
                ft[(r + hf * 8) * FSTR + n] = fmaxf(h[t][r] * sc1[t] + sh1[t], 0.f);
        }
        wave_lds_fence();

        // ---- layer 2 ----
        mlp_layer(ft, sW[1], sBias[1], lm, hf, h);
        if (PASS == 2) {
#pragma unroll
            for (int t = 0; t < 2; ++t) {
                float ls = 0.f, lq = 0.f;
#pragma unroll
                for (int r = 0; r < 8; ++r) { float v = h[t][r]; ls += v; lq += v * v; }
                accS[t] += ls; accQ[t] += lq;
            }
            continue;
        }
        // BN2 + ReLU
#pragma unroll
        for (int t = 0; t < 2; ++t) {
            int n = t * 16 + lm;
#pragma unroll
            for (int r = 0; r < 8; ++r)
                ft[(r + hf * 8) * FSTR + n] = fmaxf(h[t][r] * sc2[t] + sh2[t], 0.f);
        }
        wave_lds_fence();

        // ---- layer 3 + segment-max scatter ----
        mlp_layer(ft, sW[2], sBias[2], lm, hf, h);
#pragma unroll
        for (int t = 0; t < 2; ++t) {
            int n = t * 16 + lm;
#pragma unroll
            for (int r = 0; r < 8; ++r) {
                int dd = sdst[wv][r + hf * 8];
                atomicMax(&acc[dd * 32 + n], encf(h[t][r]));
            }
        }
    }

    if (PASS <= 2) {
#pragma unroll
        for (int t = 0; t < 2; ++t) {
            atomicAdd(&ssum[t * 16 + lm], accS[t]);
            atomicAdd(&ssq [t * 16 + lm], accQ[t]);
        }
        __syncthreads();
        if (tid < 32) {
            atomicAdd(&statOut[tid],      ssum[tid]);
            atomicAdd(&statOut[32 + tid], ssq[tid]);
        }
    }
}

__global__ void init_kernel(unsigned* __restrict__ acc, float* __restrict__ stats) {
    const long i = (long)blockIdx.x * blockDim.x + threadIdx.x;
    const long tot = (long)N_NODES * 32;
    for (long k = i; k < tot; k += (long)gridDim.x * blockDim.x) acc[k] = 0u;
    if (i < 128) stats[i] = 0.f;
}

__global__ __launch_bounds__(256) void softmax_kernel(const unsigned* __restrict__ acc,
                                                      float* __restrict__ out) {
    const int ch = blockIdx.x;          // one block per channel
    const int tid = threadIdx.x;
    __shared__ float red[256];

    float mx = -3.402823466e38f;
    for (int i = tid; i < N_NODES; i += 256)
        mx = fmaxf(mx, decf(acc[i * 32 + ch]));
    red[tid] = mx; __syncthreads();
    for (int o = 128; o > 0; o >>= 1) {
        if (tid < o) red[tid] = fmaxf(red[tid], red[tid + o]);
        __syncthreads();
    }
    mx = red[0]; __syncthreads();

    float sum = 0.f;
    for (int i = tid; i < N_NODES; i += 256)
        sum += expf(decf(acc[i * 32 + ch]) - mx);
    red[tid] = sum; __syncthreads();
    for (int o = 128; o > 0; o >>= 1) {
        if (tid < o) red[tid] += red[tid + o];
        __syncthreads();
    }
    const float inv = 1.f / red[0];

    for (int i = tid; i < N_NODES; i += 256)
        out[i * 32 + ch] = expf(decf(acc[i * 32 + ch]) - mx) * inv;
}

extern "C" void kernel_launch(void* const* d_in, const int* in_sizes, int n_in,
                              void* d_out, int out_size, void* d_ws, size_t ws_size,
                              hipStream_t stream) {
    const float* x   = (const float*)d_in[0];
    const float* pos = (const float*)d_in[1];
    const float* nrm = (const float*)d_in[2];
    const int*   ei  = (const int*)d_in[3];
    const int* esrc = ei;
    const int* edst = ei + N_EDGES;
    const float* W1 = (const float*)d_in[4];
    const float* b1 = (const float*)d_in[5];
    const float* g1 = (const float*)d_in[6];
    const float* be1 = (const float*)d_in[7];
    const float* W2 = (const float*)d_in[8];
    const float* b2 = (const float*)d_in[9];
    const float* g2 = (const float*)d_in[10];
    const float* be2 = (const float*)d_in[11];
    const float* W3 = (const float*)d_in[12];
    const float* b3 = (const float*)d_in[13];

    unsigned* acc = (unsigned*)d_ws;                                   // N*32 u32
    float* stats  = (float*)((char*)d_ws + (size_t)N_NODES * 32 * 4);  // 128 f32
    float* stat1  = stats;       // [sum1|sq1]
    float* stat2  = stats + 64;  // [sum2|sq2]
    float* out    = (float*)d_out;

    init_kernel<<<512, 256, 0, stream>>>(acc, stats);

    const int grid = 2048;  // 16384 waves, ~6 tiles each
    ppf_pass<1><<<grid, TPB, 0, stream>>>(x, pos, nrm, esrc, edst,
        W1, b1, g1, be1, W2, b2, g2, be2, W3, b3, stat1, nullptr, nullptr, nullptr);
    ppf_pass<2><<<grid, TPB, 0, stream>>>(x, pos, nrm, esrc, edst,
        W1, b1, g1, be1, W2, b2, g2, be2, W3, b3, stat2, stat1, nullptr, nullptr);
    ppf_pass<3><<<grid, TPB, 0, stream>>>(x, pos, nrm, esrc, edst,
        W1, b1, g1, be1, W2, b2, g2, be2, W3, b3, nullptr, stat1, stat2, acc);

    softmax_kernel<<<32, 256, 0, stream>>>(acc, out);
}